// CENET_86560770884087
// MI455X (gfx1250) — compile-verified
//
#include <hip/hip_runtime.h>
#include <math.h>

#define Dd   200
#define Rr   460
#define Tt   32
#define Bb   2048
#define NE   10000
#define G3   600   // 3*D

typedef __attribute__((ext_vector_type(2))) float v2f;
typedef __attribute__((ext_vector_type(8))) float v8f;

__device__ __forceinline__ float sigmoidf_(float x) { return 1.0f / (1.0f + __expf(-x)); }
__device__ __forceinline__ float logsigf_(float x) {
    // log(sigmoid(x)) = min(x,0) - log1p(exp(-|x|)), numerically stable
    return fminf(x, 0.0f) - log1pf(__expf(-fabsf(x)));
}

// ---------------------------------------------------------------------------
// Generic NT GEMM: C[M,N] = A[M,K] @ W[N,K]^T  (+ epilogue)
// 128 threads = 4 waves; each wave computes a 16(M) x 64(N) C tile using
// fp32 WMMA 16x16x4 with 4 accumulator sets sharing one A fragment per k-step
// (4x less A traffic, 4 back-to-back WMMAs on the same A operand).
// K must be a multiple of 4 (all K here: 200/400/460/600).
// EPI 0: C = acc + bias
// EPI 1: C = alpha*(acc + bias) + resid
// EPI 2: score: C = sigmoid(acc); accumulate BCE log-sigmoid terms into loss_acc
// ---------------------------------------------------------------------------
template <int EPI>
__global__ __launch_bounds__(128) void gemm_nt_wmma(
    const float* __restrict__ A, int lda,
    const float* __restrict__ W, int ldw,
    const float* __restrict__ bias,
    float* __restrict__ C, int ldc,
    int M, int N, int K,
    float alpha, const float* __restrict__ resid,
    const int* __restrict__ triples, float* __restrict__ loss_acc)
{
    const int lane = threadIdx.x & 31;   // wave32 lane
    const int wave = threadIdx.x >> 5;   // 0..3 -> M sub-tile
    const int half = lane >> 4;          // 0: lanes 0-15, 1: lanes 16-31
    const int l16  = lane & 15;
    const int m0 = ((blockIdx.y << 2) + wave) << 4;   // 16-row tile per wave
    const int n0 = blockIdx.x << 6;                   // 64-col tile per wave

    // A 16x4 frag: lanes 0-15 -> M rows, VGPR pair = K {k0+2h, k0+2h+1}
    int arow = m0 + l16; if (arow > M - 1) arow = M - 1;   // clamp ragged tiles
    const float* ap = A + (size_t)arow * lda + 2 * half;

    // 4 B fragments: N cols n0+16j+l16 (W row-major N x K)
    int coln[4];
    const float* wp[4];
#pragma unroll
    for (int j = 0; j < 4; ++j) {
        int c = n0 + j * 16 + l16;
        coln[j] = c;
        if (c > N - 1) c = N - 1;
        wp[j] = W + (size_t)c * ldw + 2 * half;
    }

    v8f acc[4];
#pragma unroll
    for (int j = 0; j < 4; ++j) acc[j] = (v8f){0.f,0.f,0.f,0.f,0.f,0.f,0.f,0.f};

#pragma unroll 2
    for (int k0 = 0; k0 < K; k0 += 4) {
        v2f a  = *(const v2f*)(ap + k0);
        v2f b0 = *(const v2f*)(wp[0] + k0);
        v2f b1 = *(const v2f*)(wp[1] + k0);
        v2f b2 = *(const v2f*)(wp[2] + k0);
        v2f b3 = *(const v2f*)(wp[3] + k0);
        acc[0] = __builtin_amdgcn_wmma_f32_16x16x4_f32(false, a, false, b0, (short)0, acc[0], false, false);
        acc[1] = __builtin_amdgcn_wmma_f32_16x16x4_f32(false, a, false, b1, (short)0, acc[1], false, false);
        acc[2] = __builtin_amdgcn_wmma_f32_16x16x4_f32(false, a, false, b2, (short)0, acc[2], false, false);
        acc[3] = __builtin_amdgcn_wmma_f32_16x16x4_f32(false, a, false, b3, (short)0, acc[3], false, false);
    }

    float lsum = 0.0f;
#pragma unroll
    for (int j = 0; j < 4; ++j) {
        const int col = coln[j];
        if (col < N) {
            float bv = 0.0f;
            if (EPI != 2 && bias) bv = bias[col];
            // C layout: VGPR r -> row m0 + r + 8*half, col fixed per lane
#pragma unroll
            for (int r = 0; r < 8; ++r) {
                int row = m0 + r + half * 8;
                if (row < M) {
                    float v = acc[j][r];
                    if (EPI == 0) {
                        C[(size_t)row * ldc + col] = v + bv;
                    } else if (EPI == 1) {
                        C[(size_t)row * ldc + col] =
                            alpha * (v + bv) + resid[(size_t)row * ldc + col];
                    } else {
                        C[(size_t)row * ldc + col] = sigmoidf_(v);
                        int obj = triples[row * 3 + 2];
                        lsum += (col == obj) ? logsigf_(v) : logsigf_(-v);
                    }
                }
            }
        }
    }
    if (EPI == 2) {
#pragma unroll
        for (int off = 16; off > 0; off >>= 1)
            lsum += __shfl_xor(lsum, off, 32);
        if (lane == 0) atomicAdd(loss_acc, lsum);
    }
}

// ---------------------------------------------------------------------------
__global__ void zero_kernel(float* __restrict__ p, int n) {
    int i = blockIdx.x * blockDim.x + threadIdx.x;
    if (i < n) p[i] = 0.0f;
}

__global__ void transpose_kernel(const float* __restrict__ in, float* __restrict__ out,
                                 int rows, int cols) {
    int i = blockIdx.x * blockDim.x + threadIdx.x;
    if (i < rows * cols) {
        int r = i / cols, c = i % cols;
        out[(size_t)c * rows + r] = in[i];
    }
}

// q_rel[b,:] = mapped_rel[rel[b],:]
__global__ void gather_rel_kernel(const float* __restrict__ mapped,
                                  const int* __restrict__ triples,
                                  float* __restrict__ q_rel) {
    int i = blockIdx.x * blockDim.x + threadIdx.x;
    if (i < Bb * Dd) {
        int b = i / Dd, d = i % Dd;
        int rel = triples[b * 3 + 1];
        q_rel[i] = mapped[(size_t)rel * Dd + d];
    }
}

// av[b,:] = aligned_all[sub[b],:] * gh_final[b,:]
__global__ void gather_mul_kernel(const float* __restrict__ aligned_all,
                                  const int* __restrict__ triples,
                                  const float* __restrict__ ghf,
                                  float* __restrict__ av) {
    int i = blockIdx.x * blockDim.x + threadIdx.x;
    if (i < Bb * Dd) {
        int b = i / Dd, d = i % Dd;
        int sub = triples[b * 3 + 0];
        av[i] = aligned_all[(size_t)sub * Dd + d] * ghf[i];
    }
}

// attn = part[t]*base; attn==0 -> -1e9; softmax over R; one block per batch row
__global__ __launch_bounds__(128) void attn_softmax_kernel(
    const float* __restrict__ part,  // [B, T*R]
    const float* __restrict__ base,  // [B, R]
    float* __restrict__ out,         // [B, R]
    int t)
{
    __shared__ float red[128];
    int b = blockIdx.x, tid = threadIdx.x;
    const float* pb = part + (size_t)b * (Tt * Rr) + (size_t)t * Rr;
    const float* bs = base + (size_t)b * Rr;

    float vals[4];
    float mx = -1.0e30f;
#pragma unroll
    for (int i = 0; i < 4; ++i) {
        int r = tid + i * 128;
        float v = -1.0e9f;
        if (r < Rr) {
            float av = pb[r] * bs[r];
            v = (av == 0.0f) ? -1.0e9f : av;
        }
        vals[i] = v;
        mx = fmaxf(mx, v);
    }
    red[tid] = mx; __syncthreads();
    for (int s = 64; s > 0; s >>= 1) {
        if (tid < s) red[tid] = fmaxf(red[tid], red[tid + s]);
        __syncthreads();
    }
    mx = red[0]; __syncthreads();

    float sum = 0.0f;
#pragma unroll
    for (int i = 0; i < 4; ++i) {
        int r = tid + i * 128;
        if (r < Rr) { vals[i] = __expf(vals[i] - mx); sum += vals[i]; }
    }
    red[tid] = sum; __syncthreads();
    for (int s = 64; s > 0; s >>= 1) {
        if (tid < s) red[tid] += red[tid + s];
        __syncthreads();
    }
    float inv = 1.0f / red[0];
#pragma unroll
    for (int i = 0; i < 4; ++i) {
        int r = tid + i * 128;
        if (r < Rr) out[(size_t)b * Rr + r] = vals[i] * inv;
    }
}

// GRU gate math. gi/gh: [B, 600] as [ir|iz|in]. If cur_ts != null, masked update.
__global__ void gru_pointwise_kernel(const float* __restrict__ gi,
                                     const float* __restrict__ gh,
                                     const float* __restrict__ hprev,
                                     float* __restrict__ hout,
                                     const int* __restrict__ cur_ts, int t) {
    int i = blockIdx.x * blockDim.x + threadIdx.x;
    if (i < Bb * Dd) {
        int b = i / Dd, d = i % Dd;
        const float* gib = gi + (size_t)b * G3;
        const float* ghb = gh + (size_t)b * G3;
        float r = sigmoidf_(gib[d]            + ghb[d]);
        float z = sigmoidf_(gib[Dd + d]       + ghb[Dd + d]);
        float n = tanhf   (gib[2 * Dd + d] + r * ghb[2 * Dd + d]);
        float hp = hprev[i];
        float hn = (1.0f - z) * n + z * hp;
        if (cur_ts) hn = (t < cur_ts[b]) ? hn : hp;
        hout[i] = hn;
    }
}

__global__ void mse_reduce_kernel(const float* __restrict__ a,
                                  const float* __restrict__ b,
                                  float* __restrict__ acc, int n) {
    __shared__ float red[256];
    int i = blockIdx.x * 256 + threadIdx.x;
    float d = 0.0f;
    if (i < n) { float x = a[i] - b[i]; d = x * x; }
    red[threadIdx.x] = d; __syncthreads();
    for (int s = 128; s > 0; s >>= 1) {
        if (threadIdx.x < s) red[threadIdx.x] += red[threadIdx.x + s];
        __syncthreads();
    }
    if (threadIdx.x == 0) atomicAdd(acc, red[0]);
}

__global__ void finalize_kernel(const float* __restrict__ acc, float* __restrict__ out) {
    if (threadIdx.x == 0) {
        out[0] = acc[0] / (float)(Bb * Dd);                 // match_loss (mean sq)
        out[1] = -acc[1] / ((float)Bb * (float)NE);         // path_loss
    }
}

// ---------------------------------------------------------------------------
// grid for 16(M)-per-wave x 64(N)-per-wave tiles, 4 waves (64 M rows) per block
static inline dim3 gblk(int m, int n) { return dim3((n + 63) / 64, (m + 63) / 64); }

extern "C" void kernel_launch(void* const* d_in, const int* in_sizes, int n_in,
                              void* d_out, int out_size, void* d_ws, size_t ws_size,
                              hipStream_t stream)
{
    const float* pre_emb = (const float*)d_in[0];
    const float* r_emb   = (const float*)d_in[1];
    const int*   triples = (const int*)d_in[2];
    const float* part    = (const float*)d_in[3];
    const int*   cur_ts  = (const int*)d_in[4];
    const float* W_map1  = (const float*)d_in[5];
    const float* b_map1  = (const float*)d_in[6];
    const float* W_map2  = (const float*)d_in[7];
    const float* b_map2  = (const float*)d_in[8];
    const float* W_attn  = (const float*)d_in[9];
    const float* b_attn  = (const float*)d_in[10];
    const float* W_ih    = (const float*)d_in[11];
    const float* W_hh    = (const float*)d_in[12];
    const float* b_ih    = (const float*)d_in[13];
    const float* b_hh    = (const float*)d_in[14];
    const float* W_h1    = (const float*)d_in[15];
    const float* b_h1    = (const float*)d_in[16];
    const float* W_h2    = (const float*)d_in[17];
    const float* b_h2    = (const float*)d_in[18];
    const float* W_align = (const float*)d_in[19];
    const float* b_align = (const float*)d_in[20];

    float* ws       = (float*)d_ws;
    float* hidden1  = ws;                      // 460*400
    float* mapped   = hidden1 + 460 * 400;     // 460*200
    float* mappedT  = mapped  + 460 * Dd;      // 200*460
    float* q_rel    = mappedT + Dd * Rr;       // B*D
    float* tmpA     = q_rel   + Bb * Dd;       // B*D
    float* base     = tmpA    + Bb * Dd;       // B*R
    float* a_soft   = base    + Bb * Rr;       // B*R
    float* xbuf     = a_soft  + Bb * Rr;       // B*D
    float* gi       = xbuf    + Bb * Dd;       // B*600
    float* ghb      = gi      + Bb * G3;       // B*600
    float* hbuf     = ghb     + Bb * G3;       // B*D
    float* tmph     = hbuf    + Bb * Dd;       // B*D
    float* pred     = tmph    + Bb * Dd;       // B*D
    float* ghf      = pred    + Bb * Dd;       // B*D
    float* alignedA = ghf     + Bb * Dd;       // NE*D
    float* av       = alignedA + (size_t)NE * Dd; // B*D
    float* accs     = av      + Bb * Dd;       // 2 scalars

    float* out   = (float*)d_out;
    float* score = out + 2;                    // [B, NE]

    const dim3 blk(128);
    const int BD  = Bb * Dd;
    const int bd_blk = (BD + 255) / 256;

    // init accumulators + h0
    zero_kernel<<<1, 256, 0, stream>>>(accs, 2);
    zero_kernel<<<bd_blk, 256, 0, stream>>>(hbuf, BD);

    // mapped_rel = (r_emb @ W_map1^T + b_map1) @ W_map2^T + b_map2
    gemm_nt_wmma<0><<<gblk(Rr, 2 * Dd), blk, 0, stream>>>(
        r_emb, Dd, W_map1, Dd, b_map1, hidden1, 2 * Dd, Rr, 2 * Dd, Dd,
        1.0f, nullptr, nullptr, nullptr);
    gemm_nt_wmma<0><<<gblk(Rr, Dd), blk, 0, stream>>>(
        hidden1, 2 * Dd, W_map2, 2 * Dd, b_map2, mapped, Dd, Rr, Dd, 2 * Dd,
        1.0f, nullptr, nullptr, nullptr);
    transpose_kernel<<<(Rr * Dd + 255) / 256, 256, 0, stream>>>(mapped, mappedT, Rr, Dd);

    // q_rel gather; base = (q_rel @ W_attn^T + b_attn) @ mapped_rel^T
    gather_rel_kernel<<<bd_blk, 256, 0, stream>>>(mapped, triples, q_rel);
    gemm_nt_wmma<0><<<gblk(Bb, Dd), blk, 0, stream>>>(
        q_rel, Dd, W_attn, Dd, b_attn, tmpA, Dd, Bb, Dd, Dd,
        1.0f, nullptr, nullptr, nullptr);
    gemm_nt_wmma<0><<<gblk(Bb, Rr), blk, 0, stream>>>(
        tmpA, Dd, mapped, Dd, nullptr, base, Rr, Bb, Rr, Dd,
        1.0f, nullptr, nullptr, nullptr);

    // GRU scan over T steps
    for (int t = 0; t < Tt; ++t) {
        attn_softmax_kernel<<<Bb, 128, 0, stream>>>(part, base, a_soft, t);
        // x = a @ mapped_rel   (as a @ mappedT^T)
        gemm_nt_wmma<0><<<gblk(Bb, Dd), blk, 0, stream>>>(
            a_soft, Rr, mappedT, Rr, nullptr, xbuf, Dd, Bb, Dd, Rr,
            1.0f, nullptr, nullptr, nullptr);
        gemm_nt_wmma<0><<<gblk(Bb, G3), blk, 0, stream>>>(
            xbuf, Dd, W_ih, Dd, b_ih, gi, G3, Bb, G3, Dd,
            1.0f, nullptr, nullptr, nullptr);
        gemm_nt_wmma<0><<<gblk(Bb, G3), blk, 0, stream>>>(
            hbuf, Dd, W_hh, Dd, b_hh, ghb, G3, Bb, G3, Dd,
            1.0f, nullptr, nullptr, nullptr);
        gru_pointwise_kernel<<<bd_blk, 256, 0, stream>>>(gi, ghb, hbuf, hbuf, cur_ts, t);
    }

    // predicted_hist = 0.1*((q_rel@W_h1^T+b_h1)@W_h2^T+b_h2) + q_rel
    gemm_nt_wmma<0><<<gblk(Bb, Dd), blk, 0, stream>>>(
        q_rel, Dd, W_h1, Dd, b_h1, tmph, Dd, Bb, Dd, Dd,
        1.0f, nullptr, nullptr, nullptr);
    gemm_nt_wmma<1><<<gblk(Bb, Dd), blk, 0, stream>>>(
        tmph, Dd, W_h2, Dd, b_h2, pred, Dd, Bb, Dd, Dd,
        0.1f, q_rel, nullptr, nullptr);

    // match_loss = mean((pred - h)^2)
    mse_reduce_kernel<<<(BD + 255) / 256, 256, 0, stream>>>(pred, hbuf, accs, BD);

    // gh = GRU(q_rel, predicted_hist)  (unmasked)
    gemm_nt_wmma<0><<<gblk(Bb, G3), blk, 0, stream>>>(
        q_rel, Dd, W_ih, Dd, b_ih, gi, G3, Bb, G3, Dd,
        1.0f, nullptr, nullptr, nullptr);
    gemm_nt_wmma<0><<<gblk(Bb, G3), blk, 0, stream>>>(
        pred, Dd, W_hh, Dd, b_hh, ghb, G3, Bb, G3, Dd,
        1.0f, nullptr, nullptr, nullptr);
    gru_pointwise_kernel<<<bd_blk, 256, 0, stream>>>(gi, ghb, pred, ghf, nullptr, 0);

    // aligned_all = pre_emb @ W_align^T + b_align; aligned_sub = gather rows
    gemm_nt_wmma<0><<<gblk(NE, Dd), blk, 0, stream>>>(
        pre_emb, Dd, W_align, Dd, b_align, alignedA, Dd, NE, Dd, Dd,
        1.0f, nullptr, nullptr, nullptr);
    gather_mul_kernel<<<bd_blk, 256, 0, stream>>>(alignedA, triples, ghf, av);

    // score_ = (aligned_sub * gh) @ aligned_all^T ; fused sigmoid + BCE loss
    gemm_nt_wmma<2><<<gblk(Bb, NE), blk, 0, stream>>>(
        av, Dd, alignedA, Dd, nullptr, score, NE, Bb, NE, Dd,
        1.0f, nullptr, triples, accs + 1);

    finalize_kernel<<<1, 32, 0, stream>>>(accs, out);
}